// Net_32444182954492
// MI455X (gfx1250) — compile-verified
//
#include <hip/hip_runtime.h>

#define NN 60000
#define RR 70
#define BB 30
#define EE 1920000
#define MCOLS (NN * 32)   // 1,920,000 flattened (n,o) columns, divisible by 16

typedef float v2f __attribute__((ext_vector_type(2)));
typedef float v8f __attribute__((ext_vector_type(8)));

__device__ __forceinline__ void atomAddF(float* p, float v) {
    unsafeAtomicAdd(p, v);   // hardware global_atomic_add_f32 on gfx1250
}

// -------- degree count (scatter-mean denominator, shared by all 3 layers) ----
__global__ __launch_bounds__(256) void k_degree(const int* __restrict__ ei,
                                                float* __restrict__ cnt) {
    int e = blockIdx.x * 256 + threadIdx.x;
    if (e < EE) atomAddF(cnt + ei[EE + e], 1.0f);
}

// -------- layer-1 weight GEMM: w1[r,c] = sum_b att1[r,b] * basis1[b,c] -------
// C tile 16x16 via V_WMMA_F32_16X16X4_F32, K=30 padded to 32 (8 k-steps).
__global__ __launch_bounds__(256) void k_weight1_gemm(
    const float* __restrict__ att1,    // [70,30]
    const float* __restrict__ basis,   // [30, MCOLS]
    float* __restrict__ w1,            // [rEnd-r0, MCOLS]
    int r0, int rEnd)
{
    __shared__ float attPad[80 * 32];  // zero-padded A
    for (int i = threadIdx.x; i < 80 * 32; i += 256) {
        int r = i >> 5, k = i & 31;
        attPad[i] = (r < RR && k < BB) ? att1[r * BB + k] : 0.0f;
    }
    __syncthreads();

    const int lane = threadIdx.x & 31;
    const int half = lane >> 4;        // 0: lanes 0-15, 1: lanes 16-31
    const int ln   = lane & 15;
    const int rowBase = r0 + blockIdx.y * 16;

    // A fragments for all 8 k-steps (ISA 16x4 f32 A layout)
    v2f a[8];
    const int m = rowBase + ln;
    #pragma unroll
    for (int s = 0; s < 8; ++s) {
        int k0 = s * 4 + half * 2;
        float a0 = (m < 80) ? attPad[m * 32 + k0]     : 0.0f;
        float a1 = (m < 80) ? attPad[m * 32 + k0 + 1] : 0.0f;
        a[s].x = a0; a[s].y = a1;
    }

    const int nColTiles = MCOLS / 16;                        // 120000
    const int waveId = blockIdx.x * 8 + (threadIdx.x >> 5);
    const int nWaves = gridDim.x * 8;

    for (int t = waveId; t < nColTiles; t += nWaves) {
        const int col = t * 16 + ln;
        v8f c = {0.f, 0.f, 0.f, 0.f, 0.f, 0.f, 0.f, 0.f};
        #pragma unroll
        for (int s = 0; s < 8; ++s) {
            int k0 = s * 4 + half * 2;
            v2f b;
            b.x = (k0     < BB) ? basis[(size_t)k0       * MCOLS + col] : 0.0f;
            b.y = (k0 + 1 < BB) ? basis[(size_t)(k0 + 1) * MCOLS + col] : 0.0f;
            c = __builtin_amdgcn_wmma_f32_16x16x4_f32(
                    false, a[s], false, b, (short)0, c, false, false);
        }
        #pragma unroll
        for (int j = 0; j < 8; ++j) {
            int r = rowBase + j + half * 8;                  // ISA C layout
            if (r < rEnd)
                w1[(size_t)(r - r0) * MCOLS + t * 16 + ln] = c[j];
        }
    }
}

// -------- layer-1 edge scatter: aggr1[dst,:] += w1[type, src, :] -------------
__global__ __launch_bounds__(256) void k_scatter1(
    const int* __restrict__ ei, const int* __restrict__ et,
    const float* __restrict__ w1, float* __restrict__ aggr1,
    int r0, int rEnd)
{
    int e = blockIdx.x * 256 + threadIdx.x;
    if (e >= EE) return;
    int t = et[e];
    if (t < r0 || t >= rEnd) return;
    int s = ei[e], d = ei[EE + e];
    const float4* row = (const float4*)(w1 + ((size_t)(t - r0) * NN + s) * 32);
    float* out = aggr1 + (size_t)d * 32;
    #pragma unroll
    for (int q = 0; q < 8; ++q) {
        float4 v = row[q];
        atomAddF(out + q * 4 + 0, v.x);
        atomAddF(out + q * 4 + 1, v.y);
        atomAddF(out + q * 4 + 2, v.z);
        atomAddF(out + q * 4 + 3, v.w);
    }
}

// -------- layer-1 finalize: x1 = relu(aggr1/cnt + root1 + bias1) -------------
__global__ __launch_bounds__(256) void k_fin1(
    const float* __restrict__ aggr1, const float* __restrict__ cnt,
    const float* __restrict__ root1, const float* __restrict__ bias1,
    float* __restrict__ x1)
{
    int i = blockIdx.x * 256 + threadIdx.x;
    if (i >= NN * 32) return;
    int n = i >> 5, o = i & 31;
    float c = fmaxf(cnt[n], 1.0f);
    x1[i] = fmaxf(aggr1[i] / c + root1[i] + bias1[o], 0.0f);
}

// -------- layer-2 weights: w2[r,i,o] = sum_b att2[r,b]*basis2[b,i,o] ---------
__global__ __launch_bounds__(256) void k_w2(
    const float* __restrict__ att2, const float* __restrict__ basis2,
    float* __restrict__ w2)
{
    int i = blockIdx.x * 256 + threadIdx.x;
    if (i >= RR * 256) return;
    int r = i >> 8, io = i & 255;
    float s = 0.f;
    for (int b = 0; b < BB; ++b) s += att2[r * BB + b] * basis2[b * 256 + io];
    w2[i] = s;
}

// -------- layer-2 edge scatter: aggr2[dst,:] += x1[src,:] @ w2[type] ---------
__global__ __launch_bounds__(256) void k_scatter2(
    const int* __restrict__ ei, const int* __restrict__ et,
    const float* __restrict__ x1, const float* __restrict__ w2g,
    float* __restrict__ aggr2)
{
    __shared__ float w2[RR * 256];   // 71,680 B — all 70 relation matrices
    for (int i = threadIdx.x; i < RR * 256; i += 256) w2[i] = w2g[i];
    __syncthreads();
    for (int e = blockIdx.x * 256 + threadIdx.x; e < EE; e += gridDim.x * 256) {
        int t = et[e], s = ei[e], d = ei[EE + e];
        const float4* xr = (const float4*)(x1 + (size_t)s * 32);
        float x[32];
        #pragma unroll
        for (int q = 0; q < 8; ++q) {
            float4 v = xr[q];
            x[q*4] = v.x; x[q*4+1] = v.y; x[q*4+2] = v.z; x[q*4+3] = v.w;
        }
        const float* wt = &w2[t * 256];
        float acc[8] = {0,0,0,0,0,0,0,0};
        #pragma unroll
        for (int i = 0; i < 32; ++i)
            #pragma unroll
            for (int o = 0; o < 8; ++o) acc[o] += x[i] * wt[i * 8 + o];
        float* out = aggr2 + (size_t)d * 8;
        #pragma unroll
        for (int o = 0; o < 8; ++o) atomAddF(out + o, acc[o]);
    }
}

// -------- layer-2 finalize: x2 = relu(aggr2/cnt + x1@root2 + bias2) ----------
__global__ __launch_bounds__(256) void k_fin2(
    const float* __restrict__ aggr2, const float* __restrict__ cnt,
    const float* __restrict__ x1, const float* __restrict__ root2,
    const float* __restrict__ bias2, float* __restrict__ x2)
{
    __shared__ float r2[256];
    if (threadIdx.x < 256) r2[threadIdx.x] = root2[threadIdx.x];
    __syncthreads();
    int n = blockIdx.x * 256 + threadIdx.x;
    if (n >= NN) return;
    float c = fmaxf(cnt[n], 1.0f);
    const float4* xr = (const float4*)(x1 + (size_t)n * 32);
    float x[32];
    #pragma unroll
    for (int q = 0; q < 8; ++q) {
        float4 v = xr[q];
        x[q*4] = v.x; x[q*4+1] = v.y; x[q*4+2] = v.z; x[q*4+3] = v.w;
    }
    #pragma unroll
    for (int o = 0; o < 8; ++o) {
        float v = aggr2[(size_t)n * 8 + o] / c + bias2[o];
        #pragma unroll
        for (int i = 0; i < 32; ++i) v += x[i] * r2[i * 8 + o];
        x2[(size_t)n * 8 + o] = fmaxf(v, 0.0f);
    }
}

// -------- layer-3 weights: w3[r,i] = sum_b att3[r,b]*basis3[b,i] -------------
__global__ __launch_bounds__(256) void k_w3(
    const float* __restrict__ att3, const float* __restrict__ basis3,
    float* __restrict__ w3)
{
    int i = blockIdx.x * 256 + threadIdx.x;
    if (i >= RR * 8) return;
    int r = i >> 3, o = i & 7;
    float s = 0.f;
    for (int b = 0; b < BB; ++b) s += att3[r * BB + b] * basis3[b * 8 + o];
    w3[i] = s;
}

// -------- layer-3 edge scatter: aggr3[dst] += dot(x2[src], w3[type]) ---------
__global__ __launch_bounds__(256) void k_scatter3(
    const int* __restrict__ ei, const int* __restrict__ et,
    const float* __restrict__ x2, const float* __restrict__ w3g,
    float* __restrict__ aggr3)
{
    __shared__ float w3[RR * 8];
    for (int i = threadIdx.x; i < RR * 8; i += 256) w3[i] = w3g[i];
    __syncthreads();
    for (int e = blockIdx.x * 256 + threadIdx.x; e < EE; e += gridDim.x * 256) {
        int t = et[e], s = ei[e], d = ei[EE + e];
        const float4* xr = (const float4*)(x2 + (size_t)s * 8);
        float4 v0 = xr[0], v1 = xr[1];
        const float* wt = &w3[t * 8];
        float m = v0.x*wt[0] + v0.y*wt[1] + v0.z*wt[2] + v0.w*wt[3]
                + v1.x*wt[4] + v1.y*wt[5] + v1.z*wt[6] + v1.w*wt[7];
        atomAddF(aggr3 + d, m);
    }
}

// -------- layer-3 finalize: out = aggr3/cnt + x2@root3 + bias3 ---------------
__global__ __launch_bounds__(256) void k_fin3(
    const float* __restrict__ aggr3, const float* __restrict__ cnt,
    const float* __restrict__ x2, const float* __restrict__ root3,
    const float* __restrict__ bias3, float* __restrict__ out)
{
    int n = blockIdx.x * 256 + threadIdx.x;
    if (n >= NN) return;
    float c = fmaxf(cnt[n], 1.0f);
    const float* x = x2 + (size_t)n * 8;
    float v = aggr3[n] / c + bias3[0];
    #pragma unroll
    for (int i = 0; i < 8; ++i) v += x[i] * root3[i];
    out[n] = v;
}

extern "C" void kernel_launch(void* const* d_in, const int* in_sizes, int n_in,
                              void* d_out, int out_size, void* d_ws, size_t ws_size,
                              hipStream_t stream)
{
    const int*   ei     = (const int*)d_in[0];
    const int*   et     = (const int*)d_in[1];
    const float* att1   = (const float*)d_in[2];
    const float* basis1 = (const float*)d_in[3];
    const float* root1  = (const float*)d_in[4];
    const float* bias1  = (const float*)d_in[5];
    const float* att2   = (const float*)d_in[6];
    const float* basis2 = (const float*)d_in[7];
    const float* root2  = (const float*)d_in[8];
    const float* bias2  = (const float*)d_in[9];
    const float* att3   = (const float*)d_in[10];
    const float* basis3 = (const float*)d_in[11];
    const float* root3  = (const float*)d_in[12];
    const float* bias3  = (const float*)d_in[13];
    float* out = (float*)d_out;

    char* ws = (char*)d_ws;
    size_t off = 0;
    auto alloc = [&](size_t bytes) {
        size_t p = off; off += (bytes + 255) & ~(size_t)255; return p;
    };
    // zeroed region (contiguous, one memset)
    size_t o_cnt   = alloc((size_t)NN * 4);
    size_t o_aggr1 = alloc((size_t)NN * 32 * 4);
    size_t o_aggr2 = alloc((size_t)NN * 8 * 4);
    size_t o_aggr3 = alloc((size_t)NN * 4);
    size_t zeroEnd = off;
    // non-zeroed scratch
    size_t o_x1 = alloc((size_t)NN * 32 * 4);
    size_t o_x2 = alloc((size_t)NN * 8 * 4);
    size_t o_w2 = alloc((size_t)RR * 256 * 4);
    size_t o_w3 = alloc((size_t)RR * 8 * 4);
    size_t fixedEnd = off;
    size_t o_w1 = off;

    // relation chunk size for the 537 MB weight1 buffer, adapted to ws_size
    size_t perRel = (size_t)NN * 32 * 4;
    size_t avail  = (ws_size > fixedEnd) ? (ws_size - fixedEnd) : 0;
    int Rc = (int)(avail / perRel);
    if (Rc > RR) Rc = RR;
    if (Rc < 1)  Rc = 1;

    float* cnt   = (float*)(ws + o_cnt);
    float* aggr1 = (float*)(ws + o_aggr1);
    float* aggr2 = (float*)(ws + o_aggr2);
    float* aggr3 = (float*)(ws + o_aggr3);
    float* x1    = (float*)(ws + o_x1);
    float* x2    = (float*)(ws + o_x2);
    float* w2    = (float*)(ws + o_w2);
    float* w3    = (float*)(ws + o_w3);
    float* w1    = (float*)(ws + o_w1);

    hipMemsetAsync(ws, 0, zeroEnd, stream);

    k_degree<<<(EE + 255) / 256, 256, 0, stream>>>(ei, cnt);

    // ---- layer 1: chunked WMMA GEMM + gather/scatter ----
    for (int r0 = 0; r0 < RR; r0 += Rc) {
        int rEnd = r0 + Rc; if (rEnd > RR) rEnd = RR;
        int rowTiles = (rEnd - r0 + 15) / 16;
        dim3 g(512, rowTiles);
        k_weight1_gemm<<<g, 256, 0, stream>>>(att1, basis1, w1, r0, rEnd);
        k_scatter1<<<(EE + 255) / 256, 256, 0, stream>>>(ei, et, w1, aggr1, r0, rEnd);
    }
    k_fin1<<<(NN * 32 + 255) / 256, 256, 0, stream>>>(aggr1, cnt, root1, bias1, x1);

    // ---- layer 2 ----
    k_w2<<<(RR * 256 + 255) / 256, 256, 0, stream>>>(att2, basis2, w2);
    k_scatter2<<<1024, 256, 0, stream>>>(ei, et, x1, w2, aggr2);
    k_fin2<<<(NN + 255) / 256, 256, 0, stream>>>(aggr2, cnt, x1, root2, bias2, x2);

    // ---- layer 3 ----
    k_w3<<<(RR * 8 + 255) / 256, 256, 0, stream>>>(att3, basis3, w3);
    k_scatter3<<<1024, 256, 0, stream>>>(ei, et, x2, w3, aggr3);
    k_fin3<<<(NN + 255) / 256, 256, 0, stream>>>(aggr3, cnt, x2, root3, bias3, out);
}